// CLRNet_Loss_39427799777622
// MI455X (gfx1250) — compile-verified
//
#include <hip/hip_runtime.h>
#include <math.h>

#define NANCH   1000
#define NFEAT   78
#define KMAX    64
#define BLK     256
#define EPSN    1e-12f
#define TWO_R_N 2160.0f   // 2 * RADIUS * NUM_POINTS = 2*15*72
#define FBIG    3.402823466e+38f

typedef float v2f __attribute__((ext_vector_type(2)));
typedef float v8f __attribute__((ext_vector_type(8)));

__device__ __forceinline__ float liou_from_S(float S) {
    // 1 - (2160 - S) / ((2160 + S) + 1e-9)
    return 1.0f - (TWO_R_N - S) / ((TWO_R_N + S) + 1e-9f);
}

__global__ __launch_bounds__(BLK) void clrnet_pair_kernel(
    const float* __restrict__ output,   // (64,1000,78)
    const float* __restrict__ label,    // (64,16,78)
    float* __restrict__ ws)             // [0..1023]=sl1l [1024..]=ll [2048..]=fl
{
    const int pair = blockIdx.x;        // 0..1023
    const int b    = pair >> 4;
    const float* __restrict__ F = output + (size_t)b * NANCH * NFEAT;
    const float* __restrict__ P = label  + (size_t)pair * NFEAT;

    __shared__ float sS[1024];     // staged f0, then S (L1 sum over 72 pts)
    __shared__ float sXR[1024];    // staged f1, then xy distance
    __shared__ float sTR[1024];    // theta delta
    __shared__ float sFOC[1024];   // focal cost
    __shared__ float sCOST[1024];
    __shared__ float sPRI[NFEAT];
    __shared__ float sRed[BLK];
    __shared__ int   sRedI[BLK];
    __shared__ float sScal[16];
    __shared__ int   sSel[KMAX];
    __shared__ float sAcc[8];

    const int tid = threadIdx.x;

    for (int j = tid; j < NFEAT; j += BLK) sPRI[j] = P[j];
    // stage logits columns 0/1 (padded region -> 0, never read)
    for (int i = tid; i < 1024; i += BLK) {
        float f0 = 0.0f, f1 = 0.0f;
        if (i < NANCH) { f0 = F[i * NFEAT + 0]; f1 = F[i * NFEAT + 1]; }
        sS[i] = f0; sXR[i] = f1;
    }
    __syncthreads();

    // ---- pass 0: column-wise log-softmax stats over the 1000 anchors ----
    float m0 = -FBIG, m1 = -FBIG;
    for (int i = tid; i < NANCH; i += BLK) {
        m0 = fmaxf(m0, sS[i]); m1 = fmaxf(m1, sXR[i]);
    }
    sRed[tid] = m0; __syncthreads();
    for (int s = BLK / 2; s > 0; s >>= 1) {
        if (tid < s) sRed[tid] = fmaxf(sRed[tid], sRed[tid + s]);
        __syncthreads();
    }
    if (tid == 0) sScal[0] = sRed[0];
    __syncthreads();
    sRed[tid] = m1; __syncthreads();
    for (int s = BLK / 2; s > 0; s >>= 1) {
        if (tid < s) sRed[tid] = fmaxf(sRed[tid], sRed[tid + s]);
        __syncthreads();
    }
    if (tid == 0) sScal[1] = sRed[0];
    __syncthreads();

    const float M0 = sScal[0], M1 = sScal[1];
    float e0 = 0.0f, e1 = 0.0f;
    for (int i = tid; i < NANCH; i += BLK) {
        e0 += expf(sS[i] - M0);
        e1 += expf(sXR[i] - M1);
    }
    sRed[tid] = e0; __syncthreads();
    for (int s = BLK / 2; s > 0; s >>= 1) {
        if (tid < s) sRed[tid] += sRed[tid + s];
        __syncthreads();
    }
    if (tid == 0) sScal[2] = M0 + logf(sRed[0]);
    __syncthreads();
    sRed[tid] = e1; __syncthreads();
    for (int s = BLK / 2; s > 0; s >>= 1) {
        if (tid < s) sRed[tid] += sRed[tid + s];
        __syncthreads();
    }
    if (tid == 0) sScal[3] = M1 + logf(sRed[0]);
    __syncthreads();

    const float logZ0 = sScal[2], logZ1 = sScal[3];
    const float p0 = sPRI[0], p1 = sPRI[1];

    // ---- pass 1: per-anchor raw quantities ----
    for (int i = tid; i < 1024; i += BLK) {
        if (i < NANCH) {
            float ls0 = sS[i]  - logZ0;
            float ls1 = sXR[i] - logZ1;
            float s0 = expf(ls0), s1 = expf(ls1);
            float foc = (1.0f - s0) * (1.0f - s0) * ls0 * p0
                      + (1.0f - s1) * (1.0f - s1) * ls1 * p1;
            const float* fr = F + i * NFEAT;
            float dx = fr[3] - sPRI[3];
            float dy = fr[4] - sPRI[4];
            float xr = sqrtf(dx * dx + dy * dy);
            float tr = fr[5] - sPRI[5];
            float S = 0.0f;
            #pragma unroll 8
            for (int j = 6; j < NFEAT; ++j) S += fabsf(fr[j] - sPRI[j]);
            sS[i] = S; sXR[i] = xr; sTR[i] = tr; sFOC[i] = foc;
        } else {
            sS[i] = 0.0f; sXR[i] = 0.0f; sTR[i] = 0.0f; sFOC[i] = 0.0f;
        }
    }
    __syncthreads();

    // ---- WMMA reduction (wave 0, EXEC all ones): four 1000-element sums ----
    // D = ones(16x4) x B(4x16) + C  => every D row holds per-column chunk sums.
    if (tid < 32) {
        const int l16 = tid & 15;
        const int hi  = tid >> 4;
        const int off = l16 * 4 + hi * 2;   // B[k,n] = vals[4n+k] lane mapping
        v2f ones = {1.0f, 1.0f};
        v8f accS = {0,0,0,0,0,0,0,0};
        v8f accX = {0,0,0,0,0,0,0,0};
        v8f accT = {0,0,0,0,0,0,0,0};
        v8f accL = {0,0,0,0,0,0,0,0};
        for (int c = 0; c < 16; ++c) {
            int base = c * 64 + off;
            float S0 = sS[base],  S1 = sS[base + 1];
            float x0 = sXR[base], x1 = sXR[base + 1];
            float t0 = sTR[base], t1 = sTR[base + 1];
            v2f bS = { S0 * S0, S1 * S1 };
            v2f bX = { x0 * x0, x1 * x1 };
            v2f bT = { t0 * t0, t1 * t1 };
            v2f bL = { (base     < NANCH) ? liou_from_S(S0) : 0.0f,
                       (base + 1 < NANCH) ? liou_from_S(S1) : 0.0f };
            accS = __builtin_amdgcn_wmma_f32_16x16x4_f32(false, ones, false, bS, (short)0, accS, false, false);
            accX = __builtin_amdgcn_wmma_f32_16x16x4_f32(false, ones, false, bX, (short)0, accX, false, false);
            accT = __builtin_amdgcn_wmma_f32_16x16x4_f32(false, ones, false, bT, (short)0, accT, false, false);
            accL = __builtin_amdgcn_wmma_f32_16x16x4_f32(false, ones, false, bL, (short)0, accL, false, false);
        }
        // D row 0 lives in VGPR0 of lanes 0..15 (N = lane)
        if (tid < 16) {
            sRed[tid]      = accS[0];
            sRed[tid + 16] = accX[0];
            sRed[tid + 32] = accT[0];
            sRed[tid + 48] = accL[0];
        }
    }
    __syncthreads();
    if (tid < 4) {
        float t = 0.0f;
        for (int j = 0; j < 16; ++j) t += sRed[tid * 16 + j];
        sScal[4 + tid] = t;   // 4:sum S^2  5:sum xr^2  6:sum tr^2  7:sum liou
    }
    __syncthreads();

    const float nd  = fmaxf(sqrtf(sScal[4]) * (1.0f / 72.0f), EPSN);
    const float nx  = fmaxf(sqrtf(sScal[5]), EPSN);
    const float nt  = fmaxf(sqrtf(sScal[6]), EPSN);
    const float inv = 1.0f / (nd * nx * nt);
    int k = (int)sScal[7];                 // truncation, matches astype(int32)
    if (k < 1) k = 1; if (k > KMAX) k = KMAX;

    // ---- pass 2: final cost ----
    for (int i = tid; i < 1024; i += BLK) {
        if (i < NANCH) {
            float q = (sS[i] * (1.0f / 72.0f)) * sXR[i] * sTR[i] * inv;
            sCOST[i] = 3.0f * q * q + sFOC[i];
        } else {
            sCOST[i] = FBIG;
        }
    }
    __syncthreads();

    // ---- selection: k smallest costs (lowest index on ties, like top_k) ----
    for (int t = 0; t < k; ++t) {
        float bv = FBIG; int bi = 0x7fffffff;
        for (int i = tid; i < NANCH; i += BLK) {
            float v = sCOST[i];
            if (v < bv) { bv = v; bi = i; }
        }
        sRed[tid] = bv; sRedI[tid] = bi; __syncthreads();
        for (int s = BLK / 2; s > 0; s >>= 1) {
            if (tid < s) {
                float ov = sRed[tid + s]; int oi = sRedI[tid + s];
                if (ov < sRed[tid] || (ov == sRed[tid] && oi < sRedI[tid])) {
                    sRed[tid] = ov; sRedI[tid] = oi;
                }
            }
            __syncthreads();
        }
        if (tid == 0) { sSel[t] = sRedI[0]; sCOST[sRedI[0]] = FBIG; }
        __syncthreads();
    }

    // ---- gather phase over the k selected anchors ----
    if (tid < 8) sAcc[tid] = 0.0f;
    __syncthreads();
    if (tid < k) {
        const float* fr = F + sSel[tid] * NFEAT;
        atomicAdd(&sAcc[0], fr[2] * fr[2]);
        atomicAdd(&sAcc[1], fr[3] * fr[3]);
        atomicAdd(&sAcc[2], fr[4] * fr[4]);
        atomicAdd(&sAcc[3], fr[5] * fr[5]);
    }
    __syncthreads();
    if (tid < 4) {
        float lab = sPRI[2 + tid];
        sScal[8 + tid] = fmaxf(sqrtf(sAcc[tid] + lab * lab), EPSN);
    }
    __syncthreads();
    if (tid < k) {
        const float* fr = F + sSel[tid] * NFEAT;
        // smooth L1 on normalized xytl
        float sl = 0.0f;
        #pragma unroll
        for (int c = 0; c < 4; ++c) {
            float den = sScal[8 + c];
            float d = fr[2 + c] / den - sPRI[2 + c] / den;
            float ad = fabsf(d);
            sl += (ad < 1.0f) ? 0.5f * d * d : (ad - 0.5f);
        }
        sl *= 0.25f;
        // per-row focal (row-wise log-softmax over 2 logits)
        float l0 = fr[0], l1 = fr[1];
        float mm = fmaxf(l0, l1);
        float lz = mm + logf(expf(l0 - mm) + expf(l1 - mm));
        int   tgt = (sPRI[1] > sPRI[0]) ? 1 : 0;
        float logpt = ((tgt == 1) ? l1 : l0) - lz;
        float pt = expf(logpt);
        float fli = -(1.0f - pt) * (1.0f - pt) * logpt;
        // liou of selected row, from cached S
        float lli = liou_from_S(sS[sSel[tid]]);
        atomicAdd(&sAcc[4], sl);
        atomicAdd(&sAcc[5], lli);
        atomicAdd(&sAcc[6], fli);
    }
    __syncthreads();
    if (tid == 0) {
        float kf = (float)k;
        ws[pair]        = sAcc[4] / kf;   // sl1l
        ws[1024 + pair] = sAcc[5] / kf;   // ll
        ws[2048 + pair] = sAcc[6] / kf;   // fl
    }
}

__global__ __launch_bounds__(BLK) void clrnet_final_kernel(
    const float* __restrict__ ws, float* __restrict__ out)
{
    __shared__ float r0[BLK], r1[BLK], r2[BLK];
    const int tid = threadIdx.x;
    float a = 0.0f, b = 0.0f, c = 0.0f;
    for (int i = tid; i < 1024; i += BLK) {
        a += ws[i]; b += ws[1024 + i]; c += ws[2048 + i];
    }
    r0[tid] = a; r1[tid] = b; r2[tid] = c; __syncthreads();
    for (int s = BLK / 2; s > 0; s >>= 1) {
        if (tid < s) { r0[tid] += r0[tid + s]; r1[tid] += r1[tid + s]; r2[tid] += r2[tid + s]; }
        __syncthreads();
    }
    if (tid == 0) {
        float sl1l = r0[0] * (1.0f / 1024.0f);
        float ll   = r1[0] * (1.0f / 1024.0f);
        float fl   = r2[0] * (1.0f / 1024.0f);
        float loss = ((sl1l > 0.0f) ? 0.5f * sl1l : 0.0f)
                   + ((ll   > 0.0f) ? 2.0f * ll   : 0.0f)
                   + ((fl   > 0.0f) ? 2.0f * fl   : 0.0f);
        out[0] = loss;
    }
}

extern "C" void kernel_launch(void* const* d_in, const int* in_sizes, int n_in,
                              void* d_out, int out_size, void* d_ws, size_t ws_size,
                              hipStream_t stream) {
    const float* output = (const float*)d_in[0];   // (64,1000,78) f32
    const float* label  = (const float*)d_in[1];   // (64,16,78)  f32
    float* ws = (float*)d_ws;                      // needs 3*1024 floats
    clrnet_pair_kernel<<<dim3(1024), dim3(BLK), 0, stream>>>(output, label, ws);
    clrnet_final_kernel<<<dim3(1), dim3(BLK), 0, stream>>>(ws, (float*)d_out);
}